// EmergentWatcher_57303453663623
// MI455X (gfx1250) — compile-verified
//
#include <hip/hip_runtime.h>

// ---------------------------------------------------------------------------
// EmergentWatcher pipeline for MI455X (gfx1250, wave32).
// Memory-bound: 768MB HBM traffic @ 23.3TB/s ~= 33us floor. The only matmul
// (cos = v_norm @ att_norm^T, [32,1024]x[1024,16]) is done with
// V_WMMA_F32_16X16X4_F32 at full fp32 precision.
// ---------------------------------------------------------------------------

typedef __attribute__((ext_vector_type(2))) float v2f;
typedef __attribute__((ext_vector_type(8))) float v8f;

#define BB 32
#define SS 2048
#define DD 1024
#define D4 (DD / 4)          // 256 float4 per row
#define KK 10
#define KP 16                // padded attractor count for the 16x16 WMMA tile
#define SCHUNK 8
#define SPER (SS / SCHUNK)   // 256 sequence positions per chunk

#define ALPHA_BASE 0.3f
#define MAX_DELTA 0.5f
#define EPS_WHITEN 1e-8f
#define EPS_NORM 1e-8f
#define EPS_FNORM 1e-12f

// workspace layout (float offsets)
#define OFF_PART  0                          // [B, SCHUNK, D] partial sums
#define OFF_VRAW  (OFF_PART + BB*SCHUNK*DD)  // [B, D]
#define OFF_VNORM (OFF_VRAW + BB*DD)         // [B, D]
#define OFF_ATT   (OFF_VNORM + BB*DD)        // [K, D]   (first normalize)
#define OFF_ATTN  (OFF_ATT + KK*DD)          // [KP, D]  (second normalize, rows 10..15 = 0)
#define OFF_COS   (OFF_ATTN + KP*DD)         // [B, KP]
#define OFF_CORR  (OFF_COS + BB*KP)          // [B, D]   v_snapped - v_raw

// --- k1: partial mean-pool sums over sequence chunks (coalesced float4) -----
__global__ void ew_k1_partial(const float* __restrict__ hs, float* __restrict__ part) {
    const int b = blockIdx.x;       // 0..31
    const int c = blockIdx.y;       // 0..SCHUNK-1
    const int t = threadIdx.x;      // 0..255 -> one float4 column group
    const float4* hs4 = reinterpret_cast<const float4*>(hs);
    float4* part4 = reinterpret_cast<float4*>(part);
    size_t base = (size_t)b * (SS * D4) + (size_t)c * SPER * D4 + t;
    float4 acc = make_float4(0.f, 0.f, 0.f, 0.f);
    for (int s = 0; s < SPER; ++s) {
        float4 v = hs4[base + (size_t)s * D4];
        acc.x += v.x; acc.y += v.y; acc.z += v.z; acc.w += v.w;
    }
    part4[(size_t)(b * SCHUNK + c) * D4 + t] = acc;
}

// --- k2: reduce chunk partials -> v_raw = mean over S -----------------------
__global__ void ew_k2_vraw(const float* __restrict__ part, float* __restrict__ vraw) {
    const int b = blockIdx.x;
    const int t = threadIdx.x;      // 0..255
    const float4* part4 = reinterpret_cast<const float4*>(part);
    float4 acc = make_float4(0.f, 0.f, 0.f, 0.f);
    for (int c = 0; c < SCHUNK; ++c) {
        float4 v = part4[(size_t)(b * SCHUNK + c) * D4 + t];
        acc.x += v.x; acc.y += v.y; acc.z += v.z; acc.w += v.w;
    }
    const float inv = 1.0f / (float)SS;
    float4 r = make_float4(acc.x * inv, acc.y * inv, acc.z * inv, acc.w * inv);
    reinterpret_cast<float4*>(vraw)[(size_t)b * D4 + t] = r;
}

// --- k3: whitening over the batch dim (ddof=0), two-pass mean/var -----------
__global__ void ew_k3_whiten(const float* __restrict__ vraw, float* __restrict__ vnorm) {
    const int d = threadIdx.x;      // 0..1023
    float x[BB];
    float sum = 0.f;
#pragma unroll
    for (int b = 0; b < BB; ++b) { x[b] = vraw[b * DD + d]; sum += x[b]; }
    const float mean = sum * (1.0f / (float)BB);
    float vs = 0.f;
#pragma unroll
    for (int b = 0; b < BB; ++b) { float t = x[b] - mean; vs += t * t; }
    const float var = vs * (1.0f / (float)BB);
    const float inv = 1.0f / sqrtf(var + EPS_WHITEN);
#pragma unroll
    for (int b = 0; b < BB; ++b) vnorm[b * DD + d] = (x[b] - mean) * inv;
}

// --- k3b: attractor L2-normalize (twice, faithful to reference) -------------
__global__ void ew_k3b_att(const float* __restrict__ attr,
                           float* __restrict__ att, float* __restrict__ attn) {
    __shared__ float red[256];
    const int k = blockIdx.x;       // 0..15
    const int t = threadIdx.x;      // 0..255
    float4* attn4 = reinterpret_cast<float4*>(attn);
    if (k >= KK) {                  // zero the WMMA pad rows
        attn4[(size_t)k * D4 + t] = make_float4(0.f, 0.f, 0.f, 0.f);
        return;
    }
    const float4 a = reinterpret_cast<const float4*>(attr)[(size_t)k * D4 + t];
    red[t] = a.x * a.x + a.y * a.y + a.z * a.z + a.w * a.w;
    __syncthreads();
    for (int off = 128; off > 0; off >>= 1) {
        if (t < off) red[t] += red[t + off];
        __syncthreads();
    }
    const float dn1 = fmaxf(sqrtf(red[0]), EPS_NORM);
    __syncthreads();
    float4 av = make_float4(a.x / dn1, a.y / dn1, a.z / dn1, a.w / dn1);
    reinterpret_cast<float4*>(att)[(size_t)k * D4 + t] = av;
    red[t] = av.x * av.x + av.y * av.y + av.z * av.z + av.w * av.w;
    __syncthreads();
    for (int off = 128; off > 0; off >>= 1) {
        if (t < off) red[t] += red[t + off];
        __syncthreads();
    }
    const float dn2 = fmaxf(sqrtf(red[0]), EPS_FNORM);
    attn4[(size_t)k * D4 + t] = make_float4(av.x / dn2, av.y / dn2, av.z / dn2, av.w / dn2);
}

// --- k4: cos = v_norm @ att_norm^T via V_WMMA_F32_16X16X4_F32 ---------------
// One block, 64 threads = 2 waves; wave w computes the 16x16 tile for batch
// rows [16w, 16w+16). EXEC is all-ones, control flow uniform (WMMA req).
// A-layout (16x4 f32): lane l holds row M=l&15, K pair {0,1} (l<16) / {2,3}.
// B-layout (4x16 f32): lane l holds col N=l&15, same K pair split.
// C/D (16x16 f32): VGPR r holds M = r + 8*(l>=16), N = l&15.
__global__ void ew_k4_cos(const float* __restrict__ vnorm,
                          const float* __restrict__ attn,
                          float* __restrict__ cosout) {
    const int lane = threadIdx.x & 31;
    const int wv = threadIdx.x >> 5;          // 0 or 1
    const int bbase = wv * 16;
    const int m = lane & 15;                  // A row / B col within tile
    const int kb = (lane >> 4) << 1;          // 0 or 2 (K pair base)
    const float* arow = vnorm + (size_t)(bbase + m) * DD + kb;
    const float* brow = attn + (size_t)m * DD + kb;

    v8f c = {};
    for (int d0 = 0; d0 < DD; d0 += 4) {
        v2f a, bm;
        a.x = arow[d0];  a.y = arow[d0 + 1];
        bm.x = brow[d0]; bm.y = brow[d0 + 1];
        c = __builtin_amdgcn_wmma_f32_16x16x4_f32(
                /*neg_a=*/false, a, /*neg_b=*/false, bm,
                /*c_mod=*/(short)0, c, /*reuse_a=*/false, /*reuse_b=*/false);
    }
#pragma unroll
    for (int r = 0; r < 8; ++r) {
        const int row = (lane < 16) ? r : (r + 8);
        cosout[(size_t)(bbase + row) * KP + (lane & 15)] = c[r];
    }
}

// --- k5: argmax (first occurrence), alpha, clamped delta -> corr ------------
__global__ void ew_k5_corr(const float* __restrict__ cosv, const float* __restrict__ att,
                           const float* __restrict__ vnorm, const float* __restrict__ vraw,
                           float* __restrict__ corr) {
    __shared__ float s_alpha;
    __shared__ int s_idx;
    const int b = blockIdx.x;
    const int t = threadIdx.x;      // 0..255
    if (t == 0) {
        float best = cosv[b * KP];
        int bi = 0;
        for (int k = 1; k < KK; ++k) {
            float v = cosv[b * KP + k];
            if (v > best) { best = v; bi = k; }   // strict > keeps first max
        }
        s_alpha = ALPHA_BASE * (1.0f - best);
        s_idx = bi;
    }
    __syncthreads();
    const float alpha = s_alpha;
    const int bi = s_idx;
    const float4 vn = reinterpret_cast<const float4*>(vnorm)[(size_t)b * D4 + t];
    const float4 vr = reinterpret_cast<const float4*>(vraw)[(size_t)b * D4 + t];
    const float4 at = reinterpret_cast<const float4*>(att)[(size_t)bi * D4 + t];
    float4 o;
    {
        float d;
        d = fminf(fmaxf(at.x - vn.x, -MAX_DELTA), MAX_DELTA); o.x = vn.x + alpha * d - vr.x;
        d = fminf(fmaxf(at.y - vn.y, -MAX_DELTA), MAX_DELTA); o.y = vn.y + alpha * d - vr.y;
        d = fminf(fmaxf(at.z - vn.z, -MAX_DELTA), MAX_DELTA); o.z = vn.z + alpha * d - vr.z;
        d = fminf(fmaxf(at.w - vn.w, -MAX_DELTA), MAX_DELTA); o.w = vn.w + alpha * d - vr.w;
    }
    reinterpret_cast<float4*>(corr)[(size_t)b * D4 + t] = o;
}

// --- k6: out = hidden_states + corr[b,:] broadcast over S (streaming) -------
__global__ void ew_k6_out(const float* __restrict__ hs, const float* __restrict__ corr,
                          float* __restrict__ out) {
    const size_t PB4 = (size_t)SS * D4;       // float4s per batch
    const size_t total4 = (size_t)BB * PB4;
    const float4* hs4 = reinterpret_cast<const float4*>(hs);
    const float4* corr4 = reinterpret_cast<const float4*>(corr);
    float4* out4 = reinterpret_cast<float4*>(out);
    const size_t stride = (size_t)gridDim.x * blockDim.x;
    for (size_t i = (size_t)blockIdx.x * blockDim.x + threadIdx.x; i < total4; i += stride) {
        const int b = (int)(i / PB4);
        const int d4 = (int)(i & (size_t)(D4 - 1));   // D4=256, layout d-fastest
        float4 h = hs4[i];
        float4 c = corr4[b * D4 + d4];                // 128KB, L2-resident
        out4[i] = make_float4(h.x + c.x, h.y + c.y, h.z + c.z, h.w + c.w);
    }
}

extern "C" void kernel_launch(void* const* d_in, const int* in_sizes, int n_in,
                              void* d_out, int out_size, void* d_ws, size_t ws_size,
                              hipStream_t stream) {
    const float* hs = (const float*)d_in[0];     // [32, 2048, 1024] fp32
    const float* attr = (const float*)d_in[1];   // [10, 1024] fp32
    float* out = (float*)d_out;                  // [32, 2048, 1024] fp32
    float* ws = (float*)d_ws;

    ew_k1_partial<<<dim3(BB, SCHUNK), 256, 0, stream>>>(hs, ws + OFF_PART);
    ew_k2_vraw<<<BB, 256, 0, stream>>>(ws + OFF_PART, ws + OFF_VRAW);
    ew_k3_whiten<<<1, 1024, 0, stream>>>(ws + OFF_VRAW, ws + OFF_VNORM);
    ew_k3b_att<<<KP, 256, 0, stream>>>(attr, ws + OFF_ATT, ws + OFF_ATTN);
    ew_k4_cos<<<1, 64, 0, stream>>>(ws + OFF_VNORM, ws + OFF_ATTN, ws + OFF_COS);
    ew_k5_corr<<<BB, 256, 0, stream>>>(ws + OFF_COS, ws + OFF_ATT,
                                       ws + OFF_VNORM, ws + OFF_VRAW, ws + OFF_CORR);
    ew_k6_out<<<4096, 256, 0, stream>>>(hs, ws + OFF_CORR, out);
}